// SparsePointFeatures_28346784153644
// MI455X (gfx1250) — compile-verified
//
#include <hip/hip_runtime.h>

// ---------------------------------------------------------------------------
// SparsePointFeatures for MI455X (gfx1250, wave32).
//
// Roofline: output = 671 MB f32 -> HBM-store bound (~29 us @ 23.3 TB/s).
// Structure:
//   k0: build the 3 WMMA B-operands in per-lane register layout (3 KB ws)
//   k1: per-batch geometry (centroid, |rel|, 3-NN density) -> 13 MB ws
//   k2: streaming expansion: 12x v_wmma_f32_16x16x4_f32 per 16-point tile,
//       LDS transpose -> contiguous 512B global_store_b128 row segments.
// ---------------------------------------------------------------------------

typedef __attribute__((ext_vector_type(2))) float v2f;
typedef __attribute__((ext_vector_type(8))) float v8f;

#define BB   32768
#define NN   20
#define D4F  64
#define BN   (BB * NN)      // 655360 points, divisible by 16
#define BIGD 1e10f

#define WPB  4              // waves per block in k2
#define LROW 132            // padded LDS row stride (floats): breaks 2-way
                            // bank conflicts (8*132 % 64 == 32), keeps 16B align

// ---------------------------------------------------------------------------
// Kernel 0: build B-tile operands in the exact per-lane WMMA layout.
// B (4x16 f32, 2 VGPRs): lanes 0-15 hold rows K0 (v.x) / K1 (v.y),
//                        lanes 16-31 hold rows K2 (v.x) / K3 (v.y).
//   rel  : rows [Wrel0, Wrel1, Wrel2, b_rel]
//   dist : rows [Wdist, 0,     b_dist, 0   ]
//   den  : rows [0,     Wden,  b_den,  0   ]
// Layout: bws[kind*128 + nb*32 + lane], kind in {rel, dist, den}, nb in 0..3.
// ---------------------------------------------------------------------------
__global__ void spf_btiles(const float* __restrict__ Wrel,
                           const float* __restrict__ brel,
                           const float* __restrict__ Wdist,
                           const float* __restrict__ bdist,
                           const float* __restrict__ Wden,
                           const float* __restrict__ bden,
                           v2f* __restrict__ bws)
{
  const int i = threadIdx.x;
  if (i >= 384) return;
  const int  kind = i >> 7;
  const int  rem  = i & 127;
  const int  nb   = rem >> 5;
  const int  lane = rem & 31;
  const int  hl   = lane & 15;
  const bool lo   = lane < 16;
  const int  col  = nb * 16 + hl;
  v2f v;
  if (kind == 0)      { v.x = lo ? Wrel[col]  : Wrel[128 + col];
                        v.y = lo ? Wrel[64 + col] : brel[col]; }
  else if (kind == 1) { v.x = lo ? Wdist[col] : bdist[col];
                        v.y = 0.0f; }
  else                { v.x = lo ? 0.0f       : bden[col];
                        v.y = lo ? Wden[col]  : 0.0f; }
  bws[i] = v;
}

// ---------------------------------------------------------------------------
// Kernel 1: per-batch geometry. One wave per batch, lane = point.
// ---------------------------------------------------------------------------
__global__ __launch_bounds__(256) void spf_geom(
    const float* __restrict__ points, const float* __restrict__ mask,
    float4* __restrict__ feat, float* __restrict__ dens,
    int* __restrict__ nvidx)
{
  const int wave = threadIdx.x >> 5;
  const int lane = threadIdx.x & 31;
  const int b    = blockIdx.x * (blockDim.x >> 5) + wave;

  float x = 0.f, y = 0.f, z = 0.f, m = 0.f;
  if (lane < NN) {
    const float* p = points + ((size_t)b * NN + lane) * 3;
    x = p[0]; y = p[1]; z = p[2];
    m = mask[(size_t)b * NN + lane];
  }

  float sx = x * m, sy = y * m, sz = z * m, sm = m;
  #pragma unroll
  for (int off = 16; off; off >>= 1) {
    sx += __shfl_xor(sx, off);
    sy += __shfl_xor(sy, off);
    sz += __shfl_xor(sz, off);
    sm += __shfl_xor(sm, off);
  }
  const float nvf = fmaxf(sm, 1.0f);
  const float rx = x - sx / nvf;
  const float ry = y - sy / nvf;
  const float rz = z - sz / nvf;
  const float cd = sqrtf(rx * rx + ry * ry + rz * rz);

  float d0 = BIGD, d1 = BIGD, d2v = BIGD;
  #pragma unroll
  for (int j = 0; j < NN; ++j) {
    const float bx = __shfl(x, j), by = __shfl(y, j);
    const float bz = __shfl(z, j), bm = __shfl(m, j);
    const float dx = x - bx, dy = y - by, dz = z - bz;
    const float dd = dx * dx + dy * dy + dz * dz;
    const bool  pv = (m > 0.f) && (bm > 0.f) && (j != lane);
    const float dv = pv ? sqrtf(fmaxf(dd, 1e-12f)) : BIGD;
    if (dv < d0)       { d2v = d1; d1 = d0; d0 = dv; }
    else if (dv < d1)  { d2v = d1; d1 = dv; }
    else if (dv < d2v) { d2v = dv; }
  }
  const int nv = (int)(sm + 0.5f);
  int k = nv - 1; if (k > 3) k = 3; if (k < 1) k = 1;
  const float acc = d0 + (k > 1 ? d1 : 0.f) + (k > 2 ? d2v : 0.f);
  const float dn  = ((m > 0.f) && (nv > 1)) ? acc / (float)k : 0.f;

  if (lane < NN) {
    const size_t p = (size_t)b * NN + lane;
    feat[p] = make_float4(rx, ry, rz, cd);
    dens[p] = dn;
  }
  if (lane == 0) {
    int idx = nv; if (idx < 1) idx = 1; if (idx > 49) idx = 49;
    nvidx[b] = idx;
  }
}

// ---------------------------------------------------------------------------
// Kernel 2: streaming feature expansion. One wave per 16-point tile.
// A (16x4 f32): lanes 0-15 -> K0,K1 ; lanes 16-31 -> K2,K3.
// D (16x16 f32, 8 VGPRs): reg r = row r (lanes 0-15) / r+8 (lanes 16-31),
// col = lane&15. Two LDS passes convert the D layout into contiguous
// 512-byte row segments emitted as global_store_b128.
// ---------------------------------------------------------------------------
__global__ __launch_bounds__(WPB * 32) void spf_wmma(
    const float4* __restrict__ feat, const float* __restrict__ dens,
    const int* __restrict__ nvidx, const float* __restrict__ emb,
    const v2f* __restrict__ bws, float* __restrict__ out)
{
  __shared__ float lds[WPB * 16 * LROW];   // 33,792 bytes

  const int wave = threadIdx.x >> 5;
  const int lane = threadIdx.x & 31;
  const unsigned tile = blockIdx.x * WPB + wave;
  const int  hl = lane & 15;
  const bool lo = lane < 16;
  float* wlds = lds + wave * 16 * LROW;

  const size_t p  = (size_t)tile * 16 + hl;
  const float4 f  = feat[p];
  const float  dn = dens[p];

  // A tiles: pass1 = [rel.x, rel.y, rel.z, 1], pass2 = [cdist, dens, 1, 0]
  v2f a1, a2;
  a1.x = lo ? f.x : f.z;  a1.y = lo ? f.y : 1.0f;
  a2.x = lo ? f.w : 1.0f; a2.y = lo ? dn  : 0.0f;

  v8f dr[4], dd[4], dv[4];
  #pragma unroll
  for (int nb = 0; nb < 4; ++nb) {
    const v2f b1  = bws[      nb * 32 + lane];
    const v2f b2d = bws[128 + nb * 32 + lane];
    const v2f b2v = bws[256 + nb * 32 + lane];
    v8f c = {};
    dr[nb] = __builtin_amdgcn_wmma_f32_16x16x4_f32(
        false, a1, false, b1,  (short)0, c, false, false);
    dd[nb] = __builtin_amdgcn_wmma_f32_16x16x4_f32(
        false, a2, false, b2d, (short)0, c, false, false);
    dv[nb] = __builtin_amdgcn_wmma_f32_16x16x4_f32(
        false, a2, false, b2v, (short)0, c, false, false);
  }

  const int rowoff = lo ? 0 : 8;

  // ---- Pass A: local cols 0..127 = [rel_f | dist_f] ----
  #pragma unroll
  for (int r = 0; r < 8; ++r) {
    const int row = r + rowoff;
    #pragma unroll
    for (int nb = 0; nb < 4; ++nb) {
      wlds[row * LROW +      nb * 16 + hl] = dr[nb][r];
      wlds[row * LROW + 64 + nb * 16 + hl] = dd[nb][r];
    }
  }
  #pragma unroll
  for (int m = 0; m < 16; ++m) {                 // LDS in-order within a wave
    const float4 v = *(const float4*)&wlds[m * LROW + lane * 4];
    const size_t prow = (size_t)tile * 16 + m;
    *(float4*)&out[prow * 256 + lane * 4] = v;   // 512B contiguous per store
  }

  // ---- Pass B: local cols 0..127 = [count_f | den_f] ----
  #pragma unroll
  for (int r = 0; r < 8; ++r) {
    const int row = r + rowoff;
    const unsigned prow = tile * 16u + row;
    const float* erow = emb + (size_t)nvidx[prow / NN] * D4F;
    #pragma unroll
    for (int nb = 0; nb < 4; ++nb) {
      wlds[row * LROW +      nb * 16 + hl] = erow[nb * 16 + hl];
      wlds[row * LROW + 64 + nb * 16 + hl] = dv[nb][r];
    }
  }
  #pragma unroll
  for (int m = 0; m < 16; ++m) {
    const float4 v = *(const float4*)&wlds[m * LROW + lane * 4];
    const size_t prow = (size_t)tile * 16 + m;
    *(float4*)&out[prow * 256 + 128 + lane * 4] = v;
  }
}

// ---------------------------------------------------------------------------
extern "C" void kernel_launch(void* const* d_in, const int* in_sizes, int n_in,
                              void* d_out, int out_size, void* d_ws, size_t ws_size,
                              hipStream_t stream) {
  const float* points = (const float*)d_in[0];
  const float* mask   = (const float*)d_in[1];
  const float* Wrel   = (const float*)d_in[2];
  const float* brel   = (const float*)d_in[3];
  const float* Wdist  = (const float*)d_in[4];
  const float* bdist  = (const float*)d_in[5];
  const float* emb    = (const float*)d_in[6];
  const float* Wden   = (const float*)d_in[7];
  const float* bden   = (const float*)d_in[8];
  float* out = (float*)d_out;

  // Workspace layout (~13.2 MB + 3 KB): feat4 | dens | nvidx | bws
  char*   ws    = (char*)d_ws;
  float4* feat  = (float4*)ws;
  float*  densp = (float*)(ws + (size_t)BN * 16);
  int*    nvidx = (int*)(ws + (size_t)BN * 20);
  v2f*    bws   = (v2f*)(ws + (size_t)BN * 20 + (size_t)BB * 4);

  spf_btiles<<<1, 384, 0, stream>>>(Wrel, brel, Wdist, bdist, Wden, bden, bws);
  spf_geom<<<BB / 8, 256, 0, stream>>>(points, mask, feat, densp, nvidx);
  spf_wmma<<<(BN / 16) / WPB, WPB * 32, 0, stream>>>(feat, densp, nvidx,
                                                     emb, bws, out);
}